// InputVector_42176578847363
// MI455X (gfx1250) — compile-verified
//
#include <hip/hip_runtime.h>

// B=128, F=1024, R=128, S=64, D=64
#define B_TOT 128
#define F_DIM 1024
#define R_DIM 128
#define S_DIM 64
#define D_DIM 64
#define LOG_2PI 1.8378770664093453f
#define MB 64     // batch rows per block (grid.y = 2)
#define XPAD 66   // s_xg row stride (dwords): even -> 8B-aligned b64, spreads banks
#define CPR 66    // coef pair-row stride (float2s): even*8B -> aligned, perturbs banks

typedef float v2f __attribute__((ext_vector_type(2)));
typedef float v8f __attribute__((ext_vector_type(8)));

// out[b,r,d] = sum_s x2[b,s]*A2[s,d] + x[b,s]*B1[s,d] + bias[d]   (per region r)
// A2 = -0.5*exp(-2*ls), B1 = exp(-2*ls)*mu, bias[d] = sum_s(A2*mu^2 - ls) - 0.5*S*log2pi
__global__ __launch_bounds__(256, 1) void region_gauss_ll_wmma(
    const float* __restrict__ x,
    const int*   __restrict__ region_idx,
    const float* __restrict__ mu,
    const float* __restrict__ log_sigma,
    float*       __restrict__ out)
{
  __shared__ int   s_idx[S_DIM];
  __shared__ float s_xg[MB][XPAD];   // gathered x rows, [b_local][s]
  // k-pair-packed coefficient planes: s_cA[p][d] = {A2[2p][d], A2[2p+1][d]}
  // -> a B fragment (rows 4ks+2h, 4ks+2h+1 at column d) is ONE contiguous b64 load
  __shared__ v2f s_cA[16][CPR];
  __shared__ v2f s_cB[16][CPR];
  __shared__ float s_bias[D_DIM];

  const int r    = blockIdx.x;       // region
  const int bb   = blockIdx.y;       // which half of the batch rows
  const int tid  = threadIdx.x;
  const int lane = tid & 31;
  const int wave = tid >> 5;         // 0..7

  // ---- stage gather indices + init bias ----
  if (tid < S_DIM) s_idx[tid]  = region_idx[r * S_DIM + tid];
  if (tid < D_DIM) s_bias[tid] = -0.5f * LOG_2PI * (float)S_DIM;
  __syncthreads();

  // warm L2/WGP$ with the far half of this region's mu/log_sigma
  __builtin_prefetch(&mu[(size_t)r * S_DIM * D_DIM + 32 * D_DIM], 0, 1);
  __builtin_prefetch(&log_sigma[(size_t)r * S_DIM * D_DIM + 32 * D_DIM], 0, 1);

  // ---- gather xg into LDS: MB*S = 4096 elements, 16 per thread ----
  #pragma unroll
  for (int i = 0; i < (MB * S_DIM) / 256; ++i) {
    int e  = tid + i * 256;
    int bl = e >> 6;                 // local batch row
    int s  = e & 63;
    s_xg[bl][s] = x[(size_t)(bb * MB + bl) * F_DIM + s_idx[s]];
  }

  // wave owns M-tile (wave&3), N-tiles { (wave>>2)*32, +16 }
  const int m0   = (wave & 3) * 16;
  const int n0   = (wave >> 2) * 32;
  const int half = lane >> 4;        // 0/1: K-pair select per ISA A/B layout
  const int lm   = lane & 15;

  v8f acc0 = {0.f,0.f,0.f,0.f,0.f,0.f,0.f,0.f};
  v8f acc1 = acc0;

  // coef-phase indices: d and pair-component are loop-invariant per thread
  const int cd  = tid & 63;          // 256 % 64 == 0
  const int cs0 = tid >> 6;          // 0..3
  const int cc  = cs0 & 1;           // component within k-pair (invariant)

  for (int h = 0; h < 2; ++h) {
    // ---- compute coefficient half s in [h*32, h*32+32): 8 rows per thread ----
    __syncthreads();                 // protects s_cA/s_cB reuse across halves
    float biasAcc = 0.f;
    #pragma unroll
    for (int i = 0; i < 8; ++i) {
      int sl = cs0 + i * 4;          // local s row 0..31
      int sp = sl >> 1;              // k-pair row
      int s  = h * 32 + sl;
      float m  = mu[((size_t)r * S_DIM + s) * D_DIM + cd];
      float ls = log_sigma[((size_t)r * S_DIM + s) * D_DIM + cd];
      float inv  = __expf(-ls);
      float inv2 = inv * inv;
      float A2   = -0.5f * inv2;
      ((float*)&s_cA[sp][cd])[cc] = A2;
      ((float*)&s_cB[sp][cd])[cc] = inv2 * m;
      biasAcc += __fmaf_rn(A2, m * m, -ls);
    }
    atomicAdd(&s_bias[cd], biasAcc); // one ds_add_f32 per thread per half
    __syncthreads();

    // ---- WMMA over this K half: 8 k-steps of 4 ----
    #pragma unroll
    for (int ks = 0; ks < 8; ++ks) {
      int kg = h * 32 + ks * 4 + 2 * half;  // global s column in s_xg
      int p  = ks * 2 + half;               // coef pair-row for this lane
      // A fragment: lane<16: K=kbase+{0,1}; lane>=16: K=kbase+{2,3}
      v2f a  = *(const v2f*)&s_xg[m0 + lm][kg];
      v2f aq; aq.x = a.x * a.x; aq.y = a.y * a.y;
      // B fragments: direct contiguous b64 loads, already in fragment order
      v2f bA0 = s_cA[p][n0 + lm];
      v2f bB0 = s_cB[p][n0 + lm];
      v2f bA1 = s_cA[p][n0 + 16 + lm];
      v2f bB1 = s_cB[p][n0 + 16 + lm];
      acc0 = __builtin_amdgcn_wmma_f32_16x16x4_f32(false, aq, false, bA0,
                                                   (short)0, acc0, false, false);
      acc0 = __builtin_amdgcn_wmma_f32_16x16x4_f32(false, a,  false, bB0,
                                                   (short)0, acc0, false, false);
      acc1 = __builtin_amdgcn_wmma_f32_16x16x4_f32(false, aq, false, bA1,
                                                   (short)0, acc1, false, false);
      acc1 = __builtin_amdgcn_wmma_f32_16x16x4_f32(false, a,  false, bB1,
                                                   (short)0, acc1, false, false);
    }
  }
  __syncthreads();                   // bias fully accumulated

  // ---- epilogue: C/D layout VGPR i, lane l -> M = i + 8*(l>=16), N = l%16 ----
  float bias0 = s_bias[n0 + lm];
  float bias1 = s_bias[n0 + 16 + lm];
  int mrow = bb * MB + m0 + 8 * half;
  #pragma unroll
  for (int i = 0; i < 8; ++i) {
    size_t base = (size_t)(mrow + i) * (R_DIM * D_DIM) + (size_t)r * D_DIM;
    out[base + n0 + lm]      = acc0[i] + bias0;
    out[base + n0 + 16 + lm] = acc1[i] + bias1;
  }
}

extern "C" void kernel_launch(void* const* d_in, const int* in_sizes, int n_in,
                              void* d_out, int out_size, void* d_ws, size_t ws_size,
                              hipStream_t stream) {
  const float* x   = (const float*)d_in[0];
  const int*   idx = (const int*)d_in[1];
  const float* mu  = (const float*)d_in[2];
  const float* ls  = (const float*)d_in[3];
  float* out = (float*)d_out;

  dim3 grid(R_DIM, B_TOT / MB);  // 128 regions x 2 batch halves = 256 blocks
  dim3 block(256);               // 8 wave32 waves
  region_gauss_ll_wmma<<<grid, block, 0, stream>>>(x, idx, mu, ls, out);
}